// PaddedToSegments_41455024341370
// MI455X (gfx1250) — compile-verified
//
#include <hip/hip_runtime.h>
#include <hip/hip_bf16.h>
#include <stdint.h>

// Reference shapes: B=16, T=2048, D=1024 (inputs [B,T,D] f32, mask [B,T] bool).
#define T_DIM 2048
#define D_DIM 1024

typedef int v4i __attribute__((ext_vector_type(4)));
typedef __attribute__((address_space(1))) v4i* gv4;   // global int4*
typedef __attribute__((address_space(3))) v4i* lv4;   // LDS int4*
typedef __attribute__((address_space(1))) char* gptr; // global char*
typedef __attribute__((address_space(3))) char* lptr; // LDS char*

// ---------------------------------------------------------------------------
// Kernel 1: stable mask scan + scatter of source row indices.
// One block, 1024 threads (32 waves). Each thread owns 32 contiguous mask
// elements; block-wide inclusive scan of per-thread counts in LDS; then each
// thread scatters its valid flat-row ids to srcidx[dest] in stable row-major
// order (matches tf.where ordering).
// ---------------------------------------------------------------------------
__global__ __launch_bounds__(1024) void pts_scan_scatter(
    const unsigned char* __restrict__ mask,  // [BT] bool, 1 byte each
    int* __restrict__ srcidx,                // [BT] compacted source row ids
    int* __restrict__ count,                 // [1]  number of valid rows
    int BT) {
  __shared__ int sums[1024];
  const int t = threadIdx.x;
  const int per = BT >> 10;        // 32 elements per thread
  const int base = t * per;

  int s = 0;
  for (int k = 0; k < per; ++k) s += (mask[base + k] != 0) ? 1 : 0;
  sums[t] = s;
  __syncthreads();

  // Inclusive Hillis-Steele scan over 1024 thread sums.
  for (int off = 1; off < 1024; off <<= 1) {
    int v = sums[t];
    int add = (t >= off) ? sums[t - off] : 0;
    __syncthreads();
    sums[t] = v + add;
    __syncthreads();
  }

  int run = (t == 0) ? 0 : sums[t - 1];    // exclusive prefix for this thread
  for (int k = 0; k < per; ++k) {
    if (mask[base + k] != 0) {
      srcidx[run++] = base + k;            // stable: row-major order preserved
    }
  }
  if (t == 1023) count[0] = sums[1023];
}

// ---------------------------------------------------------------------------
// Kernel 2: one block per output slot j (256 threads = 8 waves).
//   j <  N : async-DMA the 4KB source row global->LDS->global (16B/thread via
//            GLOBAL_LOAD_ASYNC_TO_LDS_B128 / GLOBAL_STORE_ASYNC_FROM_LDS_B128,
//            drained with s_wait_asynccnt); seg[j] = src / T.
//   j >= N : zero-fill the row; seg[j] = -1.
// Branch on j<N is block-uniform. Each thread stores exactly the LDS bytes it
// loaded, so a single per-wave s_wait_asynccnt 0 is the only sync needed.
// ---------------------------------------------------------------------------
__global__ __launch_bounds__(256) void pts_copy_rows(
    const float* __restrict__ in,      // [BT, D]
    const int* __restrict__ srcidx,    // [BT]
    const int* __restrict__ count,     // [1]
    float* __restrict__ out,           // [BT, D]
    int* __restrict__ seg,             // [BT] int32 bits in d_out tail
    int BT) {
  __shared__ alignas(16) char smem[D_DIM * 4];   // 4KB staging tile
  const int j = blockIdx.x;
  const int t = threadIdx.x;
  const int N = count[0];                        // uniform scalar load

  float* dst = out + (size_t)j * D_DIM;

  if (j < N) {
    const int src = srcidx[j];
    if (t == 0) seg[j] = src / T_DIM;            // batch index of source row

    const float* srow = in + (size_t)src * D_DIM;
    // 256 threads x 16B = 4KB row.
    gv4 g_src = (gv4)((gptr)(char*)srow) + t;
    gv4 g_dst = (gv4)((gptr)(char*)dst) + t;
    lv4 l_buf = (lv4)((lptr)smem) + t;

    // Async DMA: global -> LDS (tracked by ASYNCcnt).
    __builtin_amdgcn_global_load_async_to_lds_b128(g_src, l_buf, 0, 0);
    // Drain this wave's async loads before the store reads LDS.
    asm volatile("s_wait_asynccnt 0" ::: "memory");
    // Async DMA: LDS -> global. Implicit S_WAIT_IDLE at S_ENDPGM drains it.
    __builtin_amdgcn_global_store_async_from_lds_b128(g_dst, l_buf, 0, 0);
  } else {
    if (t == 0) seg[j] = -1;
    float4 z = make_float4(0.f, 0.f, 0.f, 0.f);
    *reinterpret_cast<float4*>(dst + t * 4) = z; // 256 x 16B = 4KB zeros
  }
}

// ---------------------------------------------------------------------------
// kernel_launch
//   d_in[0]: inputs  [B*T*D] f32
//   d_in[1]: mask    [B*T]   bool (1 byte/elem, jax bool layout)
//   d_out  : collected_values [B*T*D] f32, then segment_ids [B*T] int32
//   d_ws   : srcidx [B*T] int32 + count [1] int32  (~128KB)
// ---------------------------------------------------------------------------
extern "C" void kernel_launch(void* const* d_in, const int* in_sizes, int n_in,
                              void* d_out, int out_size, void* d_ws, size_t ws_size,
                              hipStream_t stream) {
  const float* in = (const float*)d_in[0];
  const unsigned char* mask = (const unsigned char*)d_in[1];
  const int BT = in_sizes[1];                    // 32768

  float* out = (float*)d_out;
  int* seg = (int*)(out + (size_t)BT * D_DIM);   // int32 tail of d_out

  int* srcidx = (int*)d_ws;
  int* count = srcidx + BT;

  pts_scan_scatter<<<1, 1024, 0, stream>>>(mask, srcidx, count, BT);
  pts_copy_rows<<<BT, 256, 0, stream>>>(in, srcidx, count, out, seg, BT);
}